// Prototype_loss_76398878261828
// MI455X (gfx1250) — compile-verified
//
#include <hip/hip_runtime.h>
#include <math.h>

// CDNA5 / gfx1250, wave32.
//
// loss = 0.5*mean_b( mean_d((x-p)^2) ) + 0.5*mean_b( 1 - dot/max(|x||p|,eps) )
// with p = proto[label[b]].
//
// Memory-bound: x (128MB) streamed once; protos (2MB) are L2-resident.
// Per 16-row tile (one wave): dot(x_n, p_n) = diag( X(16xD) * P^T(Dx16) ),
// accumulated with V_WMMA_F32_16X16X4_F32. The K-dimension is consumed in a
// permuted order so that every lane feeds two WMMA steps from ONE b128 load:
//   half0 lane n   : b128 @ col 8s   -> step A gets cols {8s,8s+1},   step B {8s+2,8s+3}
//   half1 lane n+16: b128 @ col 8s+4 -> step A gets cols {8s+4,8s+5}, step B {8s+6,8s+7}
// The diagonal sums over every column exactly once, so the dot is exact.
// Sum(x^2)/Sum(p^2) accumulate as element-wise v_pk_fma on the same registers.

typedef __attribute__((ext_vector_type(2))) float v2f;
typedef __attribute__((ext_vector_type(4))) float v4f;
typedef __attribute__((ext_vector_type(8))) float v8f;

#define COS_EPS 1e-8f

__global__ __launch_bounds__(256) void proto_loss_main(
    const float* __restrict__ x,
    const int*   __restrict__ label,
    const float* __restrict__ proto,
    float*       __restrict__ partials,
    int D, int ngroups)
{
  const int lane = threadIdx.x & 31;
  const int wave = threadIdx.x >> 5;
  const int group = blockIdx.x * 8 + wave;   // one 16-row tile per wave

  float contrib = 0.0f;

  if (group < ngroups) {
    const int n    = lane & 15;   // tile row owned by this lane (both halves)
    const int half = lane >> 4;   // 0 -> even 16B chunk, 1 -> odd 16B chunk
    const long long row = (long long)group * 16 + n;

    const int lab = label[row];
    const float* xr = x     + (size_t)row * D + half * 4;
    const float* pr = proto + (size_t)lab * D + half * 4;

    v8f acc0 = {}, acc1 = {};     // two 16x16 f32 accumulator chains
    v4f sxx0 = {}, sxx1 = {};     // element-wise |x|^2 partials
    v4f spp0 = {}, spp1 = {};     // element-wise |p|^2 partials

    for (int k0 = 0; k0 < D; k0 += 32) {
      // stream-ahead prefetch of x, 1KB ahead (global_prefetch_b8)
      __builtin_prefetch(xr + k0 + 256, 0, 3);
#pragma unroll
      for (int kk = 0; kk < 32; kk += 8) {
        v4f a = *reinterpret_cast<const v4f*>(xr + k0 + kk);  // b128
        v4f b = *reinterpret_cast<const v4f*>(pr + k0 + kk);  // b128
        sxx0 += a * a;                       // v_pk_fma_f32 x2, no movs
        spp0 += b * b;
        v2f alo = __builtin_shufflevector(a, a, 0, 1);
        v2f blo = __builtin_shufflevector(b, b, 0, 1);
        v2f ahi = __builtin_shufflevector(a, a, 2, 3);
        v2f bhi = __builtin_shufflevector(b, b, 2, 3);
        acc0 = __builtin_amdgcn_wmma_f32_16x16x4_f32(
            false, alo, false, blo, (short)0, acc0, false, false);
        acc1 = __builtin_amdgcn_wmma_f32_16x16x4_f32(
            false, ahi, false, bhi, (short)0, acc1, false, false);
        // rotate norm accumulators for ILP
        v4f t;
        t = sxx0; sxx0 = sxx1; sxx1 = t;
        t = spp0; spp0 = spp1; spp1 = t;
      }
    }

    v8f acc = acc0 + acc1;
    v4f sxv = sxx0 + sxx1;
    v4f spv = spp0 + spp1;
    float sxx = (sxv.x + sxv.y) + (sxv.z + sxv.w);
    float spp = (spv.x + spv.y) + (spv.z + spv.w);

    // combine the two half-lane partial sums of each row
    sxx += __shfl_xor(sxx, 16, 32);
    spp += __shfl_xor(spp, 16, 32);

    // Diagonal extraction. C/D layout: VGPR r, lanes 0-15 -> (M=r, N=lane),
    // lanes 16-31 -> (M=r+8, N=lane-16).  diag(n,n): n<8 at {vgpr n, lane n},
    // n>=8 at {vgpr n-8, lane n+16}.  Every lane selects acc[lane&7], then a
    // shuffle routes the diag value for row n=lane&15 to this lane.
    const int j = lane & 7;
    float e01 = (j & 1) ? acc[1] : acc[0];
    float e23 = (j & 1) ? acc[3] : acc[2];
    float e45 = (j & 1) ? acc[5] : acc[4];
    float e67 = (j & 1) ? acc[7] : acc[6];
    float e03 = (j & 2) ? e23 : e01;
    float e47 = (j & 2) ? e67 : e45;
    float diagv = (j & 4) ? e47 : e03;
    const int src = (n < 8) ? n : (n + 16);
    float dot = __shfl(diagv, src, 32);

    if (lane < 16) {
      // mean((x-p)^2) = (sxx - 2*dot + spp) / D
      float edu   = (sxx - 2.0f * dot + spp) * (1.0f / (float)D);
      float denom = fmaxf(sqrtf(sxx) * sqrtf(spp), COS_EPS);
      float cosv  = dot / denom;
      contrib = 0.5f * edu + 0.5f * (1.0f - cosv);
    }
    // lanes 16-31 keep contrib == 0
  }

  // wave reduction over the 16 row-contributions
  contrib += __shfl_xor(contrib, 8, 32);
  contrib += __shfl_xor(contrib, 4, 32);
  contrib += __shfl_xor(contrib, 2, 32);
  contrib += __shfl_xor(contrib, 1, 32);

  __shared__ float wsum[8];
  if (lane == 0) wsum[wave] = contrib;
  __syncthreads();
  if (threadIdx.x == 0) {
    float s = 0.0f;
#pragma unroll
    for (int i = 0; i < 8; ++i) s += wsum[i];
    partials[blockIdx.x] = s;   // deterministic: one writer per slot
  }
}

__global__ __launch_bounds__(256) void proto_loss_reduce(
    const float* __restrict__ partials, int nparts,
    float* __restrict__ out, float invB)
{
  __shared__ float sm[256];
  float s = 0.0f;
  for (int i = threadIdx.x; i < nparts; i += 256) s += partials[i];
  sm[threadIdx.x] = s;
  __syncthreads();
  for (int off = 128; off > 0; off >>= 1) {
    if (threadIdx.x < off) sm[threadIdx.x] += sm[threadIdx.x + off];
    __syncthreads();
  }
  if (threadIdx.x == 0) out[0] = sm[0] * invB;
}

extern "C" void kernel_launch(void* const* d_in, const int* in_sizes, int n_in,
                              void* d_out, int out_size, void* d_ws, size_t ws_size,
                              hipStream_t stream) {
  const float* x     = (const float*)d_in[0];  // [B, D] fp32
  const int*   label = (const int*)d_in[1];    // [B]    int32
  const float* proto = (const float*)d_in[2];  // [C, D] fp32

  const int B = in_sizes[1];
  const int D = in_sizes[0] / B;               // 512 (multiple of 32)

  const int ngroups = B / 16;                  // 16 rows per wave (B=65536 -> 4096)
  const int nblocks = (ngroups + 7) / 8;       // 8 waves per block -> 512 blocks

  float* partials = (float*)d_ws;

  proto_loss_main<<<nblocks, 256, 0, stream>>>(x, label, proto, partials, D, ngroups);
  proto_loss_reduce<<<1, 256, 0, stream>>>(partials, nblocks, (float*)d_out,
                                           1.0f / (float)B);
}